// _EMAFallbackSSM_9234179686602
// MI455X (gfx1250) — compile-verified
//
#include <hip/hip_runtime.h>

typedef __attribute__((ext_vector_type(2))) float v2f;
typedef __attribute__((ext_vector_type(8))) float v8f;

#define BB   4
#define SS   4096
#define DD   1024
#define HH   8
#define NC   16          // chunks along sequence
#define CL   (SS/NC)     // 256 steps per chunk
#define TILES (CL/16)    // 16 WMMA tiles per chunk
#define DG   (DD/16)     // 64 channel-groups per batch
#define NG   (BB*DG)     // 256 channel groups total

// lambda^e, small integer e (<=16), per-lane divergent loop is fine (reconverges before WMMA)
__device__ __forceinline__ float powi(float lam, int e) {
    float p = 1.f;
    for (int i = 0; i < e; ++i) p *= lam;
    return p;
}

// One wave handles one (16-channel group, chunk). Intra-chunk scan is done as
// Y(16x16) = M(16x16) * X(16x16) + lambda^{t+1} * carry  via 4 chained
// v_wmma_f32_16x16x4_f32 (K split into 4).
template<bool STORE>
__global__ __launch_bounds__(256)
void ema_scan_pass(const float* __restrict__ x,
                   const float* __restrict__ lambdas,
                   float* __restrict__ out,
                   float* __restrict__ wsE,        // [NG][NC][16] local chunk-end states (pass A)
                   const float* __restrict__ wsIn) // [NG][NC][16] carry-in per chunk (pass C)
{
    const int wid  = blockIdx.x * 8 + (threadIdx.x >> 5);
    const int lane = threadIdx.x & 31;
    const int g = wid >> 4;        // channel group 0..255
    const int c = wid & (NC - 1);  // chunk 0..15
    const int b  = g / DG;
    const int d0 = (g % DG) * 16;
    const float lam = lambdas[d0 >> 7];   // head = d0/128, uniform per wave
    const float oml = 1.f - lam;

    const int hi = lane >> 4;   // lane half
    const int n  = lane & 15;   // channel-in-group / row index

    // A matrix = M (scan matrix), rows t = n across lanes, K split per ISA A-layout:
    // VGPR0: K = 4k + 2*hi, VGPR1: K = 4k + 2*hi + 1
    v2f a[4];
    #pragma unroll
    for (int k = 0; k < 4; ++k) {
        const int j0 = 4 * k + 2 * hi;
        a[k].x = (j0     <= n) ? oml * powi(lam, n - j0)     : 0.f;
        a[k].y = (j0 + 1 <= n) ? oml * powi(lam, n - j0 - 1) : 0.f;
    }

    // carry factors lambda^{t+1} for D rows t = r + 8*hi
    float fc[8];
    fc[0] = powi(lam, 1 + 8 * hi);
    #pragma unroll
    for (int r = 1; r < 8; ++r) fc[r] = fc[r - 1] * lam;

    float s = STORE ? wsIn[(g * NC + c) * 16 + n] : 0.f;

    // B loads: lane reads channel d0+n, time rows (c*CL + 16*tile + 2*hi) + {0,1,4,5,8,9,12,13}
    const float* p = x + ((size_t)b * SS + (size_t)c * CL + 2 * hi) * DD + d0 + n;
    float* q = STORE ? (out + ((size_t)b * SS + (size_t)c * CL + 8 * hi) * DD + d0 + n)
                     : nullptr;

    for (int tile = 0; tile < TILES; ++tile) {
        __builtin_prefetch(p + 32 * DD, 0, 3);   // global_prefetch_b8, 2 tiles ahead

        v8f acc;
        #pragma unroll
        for (int r = 0; r < 8; ++r) acc[r] = fc[r] * s;   // seed C with carry term

        #pragma unroll
        for (int k = 0; k < 4; ++k) {
            v2f bv;
            bv.x = p[(4 * k    ) * DD];
            bv.y = p[(4 * k + 1) * DD];
            acc = __builtin_amdgcn_wmma_f32_16x16x4_f32(
                false, a[k], false, bv, (short)0, acc, false, false);
        }

        // next carry = y[15][n] : D VGPR7, lanes 16..31
        s = __shfl(acc[7], 16 + n, 32);

        if (STORE) {
            #pragma unroll
            for (int r = 0; r < 8; ++r) q[r * DD] = acc[r];
            q += 16 * DD;
        }
        p += 16 * DD;
    }

    if (!STORE) {
        if (hi == 0) wsE[(g * NC + c) * 16 + n] = s;  // local chunk-end state
    }
}

// Sequentially combine chunk carries per channel: in[c] = e[c-1] + lam^CL * in[c-1]
__global__ __launch_bounds__(256)
void ema_chunk_carry(const float* __restrict__ lambdas,
                     const float* __restrict__ wsE,
                     float* __restrict__ wsIn)
{
    const int ch = blockIdx.x * 256 + threadIdx.x;  // 0..4095
    const int g = ch >> 4, n = ch & 15;
    const int d = (g % DG) * 16 + n;
    const float lam = lambdas[d >> 7];
    float lamL = lam;                 // lam^256 via 8 squarings
    #pragma unroll
    for (int i = 0; i < 8; ++i) lamL *= lamL;

    float s = 0.f;
    for (int c = 0; c < NC; ++c) {
        wsIn[(g * NC + c) * 16 + n] = s;
        const float e = wsE[(g * NC + c) * 16 + n];
        s = e + lamL * s;
    }
}

extern "C" void kernel_launch(void* const* d_in, const int* in_sizes, int n_in,
                              void* d_out, int out_size, void* d_ws, size_t ws_size,
                              hipStream_t stream)
{
    const float* x       = (const float*)d_in[0];
    const float* lambdas = (const float*)d_in[1];
    float* out = (float*)d_out;

    float* wsE  = (float*)d_ws;              // NG*NC*16 floats = 256 KB
    float* wsIn = wsE + (size_t)NG * NC * 16;

    const int blocks = (NG * NC) / 8;        // 8 waves per 256-thread block -> 512 blocks

    ema_scan_pass<false><<<blocks, 256, 0, stream>>>(x, lambdas, nullptr, wsE, nullptr);
    ema_chunk_carry<<<(BB * DD) / 256, 256, 0, stream>>>(lambdas, wsE, wsIn);
    ema_scan_pass<true ><<<blocks, 256, 0, stream>>>(x, lambdas, out, nullptr, wsIn);
}